// BGe_31542239822244
// MI455X (gfx1250) — compile-verified
//
#include <hip/hip_runtime.h>
#include <hip/hip_bf16.h>
#include <math.h>

#ifndef M_PI
#define M_PI 3.14159265358979323846
#endif

#define D   64
#define STR 65   // LDS row stride for Cholesky matrix (pad to break 64-bank conflicts)

typedef float v2f __attribute__((ext_vector_type(2)));
typedef float v8f __attribute__((ext_vector_type(8)));

// ---------------- setup kernels ----------------

__global__ __launch_bounds__(64) void k_mean(const float* __restrict__ data,
                                             float* __restrict__ mean, int N) {
  int i = threadIdx.x;
  float s = 0.f;
  for (int n = 0; n < N; ++n) s += data[n * D + i];
  mean[i] = s / (float)N;
}

__global__ __launch_bounds__(64) void k_lgamma(float* __restrict__ lg, int N) {
  int l = threadIdx.x;
  double alpha_mu = 1.0, alpha_lambd = (double)D + 2.0;
  double small_t = alpha_mu * (alpha_lambd - D - 1.0) / (alpha_mu + 1.0);
  double v = 0.5 * (log(alpha_mu) - log((double)N + alpha_mu))
           + lgamma(0.5 * ((double)N + alpha_lambd - D + l + 1.0))
           - lgamma(0.5 * (alpha_lambd - D + l + 1.0))
           - 0.5 * (double)N * log(M_PI)
           + 0.5 * (alpha_lambd - D + 2.0 * l + 1.0) * log(small_t);
  lg[l] = (float)v;
}

// ---------------- R = small_t*I + Xc^T Xc + coef * mean mean^T ----------------
// One workgroup, 16 waves; wave t owns 16x16 output tile (ti,tj).
// K = N runs in 64-row LDS-staged chunks, 16 chained V_WMMA_F32_16X16X4_F32 per chunk.

#define KCHUNK 64
#define SSTR   66   // staging stride: rows k and k+2 start 4 banks apart

__global__ __launch_bounds__(512) void k_R(const float* __restrict__ data,
                                           const float* __restrict__ mean,
                                           float* __restrict__ R, int N) {
  __shared__ float xcS[KCHUNK * SSTR];
  int tid  = threadIdx.x;
  int wave = tid >> 5;
  int lane = tid & 31;
  int half = lane >> 4, l16 = lane & 15;
  int ti = (wave >> 2) * 16;
  int tj = (wave & 3) * 16;

  v8f acc = {0.f, 0.f, 0.f, 0.f, 0.f, 0.f, 0.f, 0.f};
  int nchunks = (N + KCHUNK - 1) / KCHUNK;
  for (int ch = 0; ch < nchunks; ++ch) {
    int n0 = ch * KCHUNK;
    // stage centered data chunk (zero-pad past N)
    for (int e = tid; e < KCHUNK * D; e += 512) {
      int rn = e >> 6, ci = e & 63;
      int n  = n0 + rn;
      xcS[rn * SSTR + ci] = (n < N) ? (data[n * D + ci] - mean[ci]) : 0.f;
    }
    __syncthreads();
#pragma unroll
    for (int q = 0; q < KCHUNK / 4; ++q) {
      int k0 = 4 * q + 2 * half;           // A layout: lanes 16-31 hold K+2
      v2f a, bb;
      a.x  = xcS[k0 * SSTR + ti + l16];
      a.y  = xcS[(k0 + 1) * SSTR + ti + l16];
      bb.x = xcS[k0 * SSTR + tj + l16];
      bb.y = xcS[(k0 + 1) * SSTR + tj + l16];
      acc = __builtin_amdgcn_wmma_f32_16x16x4_f32(
          false, a, false, bb, (short)0, acc, false, false);
    }
    __syncthreads();
  }

  const float alpha_mu = 1.0f;
  float coef    = (float)N * alpha_mu / ((float)N + alpha_mu);
  float small_t = alpha_mu * ((float)D + 2.0f - (float)D - 1.0f) / (alpha_mu + 1.0f);
#pragma unroll
  for (int v = 0; v < 8; ++v) {
    int i = ti + v + 8 * half;             // C/D layout: VGPR v -> M=v / M=v+8
    int k = tj + l16;
    R[i * D + k] = acc[v] + coef * mean[i] * mean[k] + ((i == k) ? small_t : 0.f);
  }
}

// ---------------- per-(graph,node) masked Cholesky logdet ----------------
// One wave32 per factorization. blockIdx = (b*64 + j)*2 + which,
// which==0 -> parents mask, which==1 -> parents + node j.

__global__ __launch_bounds__(32) void k_chol(const float* __restrict__ R,
                                             const int*   __restrict__ w,
                                             float*       __restrict__ ld_arr) {
  __shared__ float M[D * STR];
  __shared__ int   maskS[D];
  __shared__ float red[32];

  int blk   = blockIdx.x;
  int which = blk & 1;
  int pair  = blk >> 1;
  int j     = pair & (D - 1);
  int b     = pair >> 6;
  int lane  = threadIdx.x;

  // parent mask of node j: column j of w[b]
  for (int i = lane; i < D; i += 32) {
    int wv = w[(b * D + i) * D + j];
    maskS[i] = (wv == 1) || (which && (i == j));
  }
  __syncthreads();

  // build masked matrix (identity in masked-out rows/cols)
  for (int idx = lane; idx < D * D; idx += 32) {
    int i = idx >> 6, k = idx & 63;
    float v = (maskS[i] & maskS[k]) ? R[idx] : ((i == k) ? 1.0f : 0.0f);
    M[i * STR + k] = v;
  }
  __syncthreads();

  // blocked right-looking Cholesky, nb = 16
  int half = lane >> 4;      // 0: lanes 0-15, 1: lanes 16-31
  int l16  = lane & 15;

  for (int s = 0; s < 4; ++s) {
    int o = s * 16;

    // ---- panel factorization (columns o..o+15, rows o..63), scalar/LDS ----
    for (int k = 0; k < 16; ++k) {
      int ck = o + k;
      float dsq  = M[ck * STR + ck];
      float dv   = sqrtf(dsq);
      float invd = 1.0f / dv;
      for (int r = lane; r < D; r += 32) {
        if (r > ck)       M[r * STR + ck] *= invd;
        else if (r == ck) M[r * STR + ck]  = dv;
      }
      __syncthreads();
      for (int c2 = ck + 1; c2 < o + 16; ++c2) {
        float lc = M[c2 * STR + ck];
        for (int r = lane; r < D; r += 32)
          if (r >= c2) M[r * STR + c2] -= M[r * STR + ck] * lc;
      }
      __syncthreads();
    }

    // ---- trailing syrk update via V_WMMA_F32_16X16X4_F32 ----
    // C[ti..ti+15][tj..tj+15] -= Lpanel(ti) * Lpanel(tj)^T  (K = 16, 4 chunks of 4)
    for (int ti = o + 16; ti < D; ti += 16) {
      for (int tj = o + 16; tj <= ti; tj += 16) {
        v8f acc = {0.f, 0.f, 0.f, 0.f, 0.f, 0.f, 0.f, 0.f};
#pragma unroll
        for (int q = 0; q < 4; ++q) {
          int k0 = o + 4 * q + 2 * half;   // A layout: lanes 16-31 hold K+2
          v2f a, bb;
          a.x  = M[(ti + l16) * STR + k0];
          a.y  = M[(ti + l16) * STR + k0 + 1];
          bb.x = M[(tj + l16) * STR + k0];     // B = Lpanel(tj)^T, mirrored layout
          bb.y = M[(tj + l16) * STR + k0 + 1];
          acc = __builtin_amdgcn_wmma_f32_16x16x4_f32(
              false, a, false, bb, (short)0, acc, false, false);
        }
        // C/D layout: VGPR v -> row M=v (lanes 0-15) / M=v+8 (lanes 16-31)
#pragma unroll
        for (int v = 0; v < 8; ++v)
          M[(ti + v + 8 * half) * STR + tj + l16] -= acc[v];
        __syncthreads();
      }
    }
    __syncthreads();
  }

  // logdet = 2 * sum(log(diag(L))); masked-out rows contribute log(1)=0
  float p = logf(M[lane * STR + lane]) + logf(M[(lane + 32) * STR + (lane + 32)]);
  red[lane] = p;
  __syncthreads();
  if (lane == 0) {
    float t = 0.f;
    for (int i2 = 0; i2 < 32; ++i2) t += red[i2];
    ld_arr[blk] = 2.0f * t;
  }
}

// ---------------- recombine per-node terms ----------------

__global__ __launch_bounds__(64) void k_score(const int* __restrict__ w,
                                              const unsigned char* __restrict__ interv,
                                              const float* __restrict__ ld_arr,
                                              const float* __restrict__ lg,
                                              float* __restrict__ out, int N) {
  int b = blockIdx.x;
  int j = threadIdx.x;
  int np = 0;
  for (int i = 0; i < D; ++i) np += (w[(b * D + i) * D + j] == 1);
  float c = (float)N + 2.0f;                 // N + alpha_lambd - d
  int base = (b * D + j) * 2;
  float ld_p  = ld_arr[base];
  float ld_pj = ld_arr[base + 1];
  float term = 0.5f * ((c + (float)np) * ld_p - (c + (float)np + 1.0f) * ld_pj)
             + lg[np];
  float keep = interv[b * D + j] ? 0.f : 1.f;
  __shared__ float red[D];
  red[j] = term * keep;
  __syncthreads();
  if (j == 0) {
    float s = 0.f;
    for (int q = 0; q < D; ++q) s += red[q];
    out[b] = s;
  }
}

// ---------------- launch ----------------

extern "C" void kernel_launch(void* const* d_in, const int* in_sizes, int n_in,
                              void* d_out, int out_size, void* d_ws, size_t ws_size,
                              hipStream_t stream) {
  const float* data           = (const float*)d_in[0];
  const int*   w              = (const int*)d_in[1];
  const unsigned char* interv = (const unsigned char*)d_in[2];
  float* out = (float*)d_out;

  int N     = in_sizes[0] / D;
  int batch = in_sizes[1] / (D * D);

  float* ws     = (float*)d_ws;
  float* mean   = ws;                    // 64
  float* lg     = ws + 64;               // 64
  float* R      = ws + 128;              // 4096
  float* ld_arr = ws + 128 + 4096;       // batch*D*2 (256 KB at batch=512)

  k_mean  <<<1, 64, 0, stream>>>(data, mean, N);
  k_lgamma<<<1, 64, 0, stream>>>(lg, N);
  k_R     <<<1, 512, 0, stream>>>(data, mean, R, N);
  k_chol  <<<batch * D * 2, 32, 0, stream>>>(R, w, ld_arr);
  k_score <<<batch, 64, 0, stream>>>(w, interv, ld_arr, lg, out, N);
}